// Model_1984274891243
// MI455X (gfx1250) — compile-verified
//
#include <hip/hip_runtime.h>
#include <hip/hip_bf16.h>
#include <math.h>

typedef __attribute__((ext_vector_type(2))) float v2f;
typedef __attribute__((ext_vector_type(8))) float v8f;

// db4 filters (reference _LO / _HI); DEC_*_R[t] = *[7-t], REC_*_R[t] = *[t]
__device__ __constant__ float c_LO[8] = {
  -0.010597401784997278f,  0.032883011666982945f,  0.030841381835986965f,
  -0.18703481171888114f,  -0.02798376941698385f,   0.6308807679295904f,
   0.7148465705525415f,    0.23037781330885523f };
__device__ __constant__ float c_HI[8] = {
  -0.23037781330885523f,   0.7148465705525415f,   -0.6308807679295904f,
  -0.02798376941698385f,   0.18703481171888114f,   0.030841381835986965f,
  -0.032883011666982945f, -0.010597401784997278f };

__device__ inline v8f wmma4(v2f a, v2f b, v8f c) {
  // V_WMMA_F32_16X16X4_F32 : D = A(16x4,f32) * B(4x16,f32) + C(16x16,f32)
  return __builtin_amdgcn_wmma_f32_16x16x4_f32(false, a, false, b, (short)0, c, false, false);
}
__device__ inline v2f ld2(const float* p) { return *(const v2f*)p; }
__device__ inline void st2(float* p, v2f v) { *(v2f*)p = v; }
__device__ inline float gelu_exact(float v) {
  return 0.5f * v * (1.0f + erff(v * 0.70710678118654752440f));
}

// ---------------------------------------------------------------- utilities
__global__ __launch_bounds__(256) void zero_kernel(float* p, int n) {
  int i = blockIdx.x * 256 + threadIdx.x;
  if (i < n) p[i] = 0.0f;
}

// x_enc (8,512,32) -> xr (8,512) taking channel 0
__global__ __launch_bounds__(256) void extract_kernel(const float* __restrict__ x, float* __restrict__ xr) {
  int i = blockIdx.x * 256 + threadIdx.x;
  if (i < 8 * 512) xr[i] = x[i * 32];
}

// ---------------------------------------------------------------- DWT bands
// symmetric extension value: ext index j in [0, n+13], e(i) = ext(i+1)
__device__ inline float ext_val(const float* x, int n, int j) {
  if (j < 7) return x[6 - j];
  if (j < 7 + n) return x[j - 7];
  return x[2 * n + 6 - j];
}

__device__ void dwt_step_dev(const float* x, int n, float* a, float* d, int tid) {
  int nout = (n + 7) >> 1;
  for (int k = tid; k < nout; k += 128) {
    float la = 0.f, ld = 0.f;
#pragma unroll
    for (int t = 0; t < 8; ++t) {
      float v = ext_val(x, n, 2 * k + t + 1);
      la += v * c_LO[7 - t];
      ld += v * c_HI[7 - t];
    }
    a[k] = la; d[k] = ld;
  }
}

// out[m] (len 2n-6); a/d may be null (treated as zeros)
__device__ void idwt_dev(const float* a, const float* d, int n, float* out, int tid) {
  int outlen = 2 * n - 6;
  for (int m = tid; m < outlen; m += 128) {
    float acc = 0.f;
#pragma unroll
    for (int t = 0; t < 8; ++t) {
      int idx = m + t - 1;
      if (idx >= 0 && idx < 2 * n - 1 && !(idx & 1)) {
        int i2 = idx >> 1;
        if (a) acc += a[i2] * c_LO[t];
        if (d) acc += d[i2] * c_HI[t];
      }
    }
    out[m] = acc;
  }
}

// 32 blocks: row = blockIdx>>2, band = blockIdx&3
__global__ __launch_bounds__(128) void dwt_bands_kernel(const float* __restrict__ xr,
                                                        float* __restrict__ bands) {
  __shared__ float s_x[512];
  __shared__ float s_a1[259], s_d1[259];
  __shared__ float s_a2[133], s_d2[133];
  __shared__ float s_a3[70],  s_d3[70];
  __shared__ float s_b1[134], s_b2[260];
  const int tid = threadIdx.x;
  const int row = blockIdx.x >> 2;
  const int bd  = blockIdx.x & 3;

  for (int i = tid; i < 512; i += 128) s_x[i] = xr[row * 512 + i];
  __syncthreads();
  dwt_step_dev(s_x, 512, s_a1, s_d1, tid); __syncthreads();
  dwt_step_dev(s_a1, 259, s_a2, s_d2, tid); __syncthreads();
  dwt_step_dev(s_a2, 133, s_a3, s_d3, tid); __syncthreads();

  // reconstruction keeping only band bd: clist = [a3, d3, d2, d1]
  idwt_dev(bd == 0 ? s_a3 : nullptr, bd == 1 ? s_d3 : nullptr, 70, s_b1, tid); // ->134
  __syncthreads();
  idwt_dev(s_b1, bd == 2 ? s_d2 : nullptr, 133, s_b2, tid);                    // ->260 (trim 134->133)
  __syncthreads();
  idwt_dev(s_b2, bd == 3 ? s_d1 : nullptr, 259,                                // ->512 (trim 260->259)
           bands + (bd * 8 + row) * 512, tid);
}

// ---------------------------------------------------------------- embed + tanh
__global__ __launch_bounds__(256) void embed_kernel(const float* __restrict__ band,
                                                    const float* __restrict__ ew,
                                                    float* __restrict__ emb,
                                                    float* __restrict__ tx) {
  int i = blockIdx.x * 256 + threadIdx.x;   // < 524288
  int d = i & 127;
  int n = i >> 7;
  int l = n & 511, seq = n >> 9;
  const float* b = band + seq * 512;
  float xm = b[(l + 511) & 511], x0 = b[l], xp = b[(l + 1) & 511];
  float e = ew[d * 3] * xm + ew[d * 3 + 1] * x0 + ew[d * 3 + 2] * xp;
  emb[i] = e;
  tx[i] = tanhf(e);
}

// ---------------------------------------------------------------- Chebyshev helpers
__global__ __launch_bounds__(128) void cheb_colsum_kernel(const float* __restrict__ cheb,
                                                          int K, float* __restrict__ s0) {
  int j = threadIdx.x;                 // 128
  float s = 0.f;
  for (int d = 0; d < 128; ++d) s += cheb[(d * 128 + j) * K];
  s0[j] = s;
}

// Tnew = 2*x*Tlast - (Tpp ? Tpp : 1)
__global__ __launch_bounds__(256) void cheb_rec_kernel(const float* __restrict__ x,
                                                       const float* __restrict__ Tlast,
                                                       const float* __restrict__ Tpp,
                                                       float* __restrict__ Tnew, int n) {
  int i = blockIdx.x * 256 + threadIdx.x;
  if (i < n) {
    float p2 = Tpp ? Tpp[i] : 1.0f;
    Tnew[i] = 2.0f * x[i] * Tlast[i] - p2;
  }
}

// ---------------------------------------------------------------- WMMA GEMM: OUT(Nx128) += A(Nx128) @ W(128x128) [+colbias]
// W element (row d=k, col j) at W[d*w_rs + j*w_cs]
// LDS layouts (even stride 34 -> 8B-aligned v2f loads, single ds_load_b64 per fragment):
//   sA [row][k]   : A fragment  a = ld2(&sA[m][kbase])
//   sWt[col][k]   : B fragment  b = ld2(&sWt[col][kbase])   (k-contiguous, transposed)
__global__ __launch_bounds__(256) void gemm128_acc_kernel(const float* __restrict__ A,
                                                          const float* __restrict__ W,
                                                          int w_rs, int w_cs,
                                                          float* __restrict__ OUT,
                                                          const float* __restrict__ colbias) {
  __shared__ float sA [128][34];
  __shared__ float sWt[128][34];
  const int tid = threadIdx.x, lane = tid & 31;
  const int m0 = (tid >> 5) * 16;
  const int laneM = lane & 15;
  const int koff  = (lane >> 4) << 1;   // lanes 16-31 hold K+2/K+3
  v8f acc[8];
#pragma unroll
  for (int j = 0; j < 8; ++j)
#pragma unroll
    for (int v = 0; v < 8; ++v) acc[j][v] = 0.0f;

  const int n0 = blockIdx.x * 128;
  for (int kc = 0; kc < 128; kc += 32) {
    __syncthreads();
    // A chunk 128x32 as float2 (coalesced 8B loads)
    for (int i = tid; i < 128 * 16; i += 256) {
      int r = i >> 4, c2 = (i & 15) << 1;
      st2(&sA[r][c2], ld2(&A[(n0 + r) * 128 + kc + c2]));
    }
    // W chunk 32x128 -> transposed k-contiguous
    for (int i = tid; i < 32 * 128; i += 256) {
      int r = i >> 7, c = i & 127;
      sWt[c][r] = W[(kc + r) * w_rs + c * w_cs];
    }
    __syncthreads();
#pragma unroll
    for (int kk = 0; kk < 32; kk += 4) {
      const int kbase = kk + koff;
      v2f a = ld2(&sA[m0 + laneM][kbase]);
#pragma unroll
      for (int j = 0; j < 8; ++j) {
        v2f b = ld2(&sWt[j * 16 + laneM][kbase]);
        acc[j] = wmma4(a, b, acc[j]);
      }
    }
  }
  const int rofs = (lane >> 4) << 3;   // 0 or 8
#pragma unroll
  for (int j = 0; j < 8; ++j)
#pragma unroll
    for (int v = 0; v < 8; ++v) {
      int m = m0 + v + rofs, col = j * 16 + laneM;
      float val = acc[j][v];
      if (colbias) val += colbias[col];
      OUT[(n0 + m) * 128 + col] += val;
    }
}

// ---------------------------------------------------------------- dilated conv (3 taps) as WMMA GEMM + GELU
// mode 0: OUT = gelu(conv+bias) ; mode 1: OUT += gelu(conv+bias) + RES
__global__ __launch_bounds__(256) void conv_gelu_kernel(const float* __restrict__ IN,  // (8,512,128)
                                                        const float* __restrict__ Wc,  // (128,128,3)
                                                        const float* __restrict__ bias,
                                                        float* __restrict__ OUT,
                                                        const float* __restrict__ RES,
                                                        int dil, int mode) {
  __shared__ float sA [128][34];
  __shared__ float sWt[128][34];
  const int tid = threadIdx.x, lane = tid & 31;
  const int seq = blockIdx.x >> 2;
  const int l0  = (blockIdx.x & 3) * 128;
  const float* Abase = IN + seq * 512 * 128;
  const int m0 = (tid >> 5) * 16;
  const int laneM = lane & 15;
  const int koff  = (lane >> 4) << 1;
  v8f acc[8];
#pragma unroll
  for (int j = 0; j < 8; ++j)
#pragma unroll
    for (int v = 0; v < 8; ++v) acc[j][v] = 0.0f;

  for (int t = 0; t < 3; ++t) {
    const int s = (t - 1) * dil;
    for (int kc = 0; kc < 128; kc += 32) {
      __syncthreads();
      for (int i = tid; i < 128 * 16; i += 256) {
        int r = i >> 4, c2 = (i & 15) << 1;
        int l = l0 + r + s;
        v2f z; z.x = 0.f; z.y = 0.f;
        st2(&sA[r][c2], (l >= 0 && l < 512) ? ld2(&Abase[l * 128 + kc + c2]) : z);
      }
      for (int i = tid; i < 32 * 128; i += 256) {
        int r = i >> 7, c = i & 127;           // r=din-in-chunk, c=dout
        sWt[c][r] = Wc[c * 384 + (kc + r) * 3 + t];
      }
      __syncthreads();
#pragma unroll
      for (int kk = 0; kk < 32; kk += 4) {
        const int kbase = kk + koff;
        v2f a = ld2(&sA[m0 + laneM][kbase]);
#pragma unroll
        for (int j = 0; j < 8; ++j) {
          v2f b = ld2(&sWt[j * 16 + laneM][kbase]);
          acc[j] = wmma4(a, b, acc[j]);
        }
      }
    }
  }
  const int rofs = (lane >> 4) << 3;
#pragma unroll
  for (int j = 0; j < 8; ++j)
#pragma unroll
    for (int v = 0; v < 8; ++v) {
      int m = m0 + v + rofs, col = j * 16 + laneM;
      float g = gelu_exact(acc[j][v] + bias[col]);
      int gi = (seq * 512 + l0 + m) * 128 + col;
      if (mode == 0) OUT[gi] = g;
      else           OUT[gi] += g + RES[gi];
    }
}

// ---------------------------------------------------------------- pred: H(seq,96,128) = pred_w(96,512) @ fused_seq(512,128) + pred_b[p]
__global__ __launch_bounds__(256) void pred_gemm_kernel(const float* __restrict__ PW,
                                                        const float* __restrict__ F,
                                                        const float* __restrict__ pb,
                                                        float* __restrict__ H) {
  __shared__ float sA [96][34];
  __shared__ float sBt[128][34];
  const int tid = threadIdx.x, lane = tid & 31;
  const int seq = blockIdx.x;
  const int j0 = (tid >> 5) * 16;         // d-column tile per wave
  const int laneM = lane & 15;
  const int koff  = (lane >> 4) << 1;
  v8f acc[6];
#pragma unroll
  for (int mt = 0; mt < 6; ++mt)
#pragma unroll
    for (int v = 0; v < 8; ++v) acc[mt][v] = 0.0f;

  for (int kc = 0; kc < 512; kc += 32) {
    __syncthreads();
    for (int i = tid; i < 96 * 16; i += 256) {
      int r = i >> 4, c2 = (i & 15) << 1;
      st2(&sA[r][c2], ld2(&PW[r * 512 + kc + c2]));
    }
    for (int i = tid; i < 32 * 128; i += 256) {
      int r = i >> 7, c = i & 127;
      sBt[c][r] = F[(seq * 512 + kc + r) * 128 + c];
    }
    __syncthreads();
#pragma unroll
    for (int kk = 0; kk < 32; kk += 4) {
      const int kbase = kk + koff;
      v2f b = ld2(&sBt[j0 + laneM][kbase]);
#pragma unroll
      for (int mt = 0; mt < 6; ++mt) {
        v2f a = ld2(&sA[mt * 16 + laneM][kbase]);
        acc[mt] = wmma4(a, b, acc[mt]);
      }
    }
  }
  const int rofs = (lane >> 4) << 3;
#pragma unroll
  for (int mt = 0; mt < 6; ++mt)
#pragma unroll
    for (int v = 0; v < 8; ++v) {
      int p = mt * 16 + v + rofs, d = j0 + laneM;
      H[(seq * 96 + p) * 128 + d] = acc[mt][v] + pb[p];
    }
}

// ---------------------------------------------------------------- proj: out(8,96,3)
__global__ __launch_bounds__(256) void proj_kernel(const float* __restrict__ H,
                                                   const float* __restrict__ pw,
                                                   const float* __restrict__ pb,
                                                   float* __restrict__ out) {
  int i = blockIdx.x * 256 + threadIdx.x;
  if (i >= 8 * 96 * 3) return;
  int q = i % 3, np = i / 3;
  float s = pb[q];
  const float* h = H + np * 128;
  const float* w = pw + q * 128;
  for (int d = 0; d < 128; ++d) s += h[d] * w[d];
  out[i] = s;
}

// ---------------------------------------------------------------- launch
extern "C" void kernel_launch(void* const* d_in, const int* in_sizes, int n_in,
                              void* d_out, int out_size, void* d_ws, size_t ws_size,
                              hipStream_t stream) {
  (void)in_sizes; (void)n_in; (void)out_size; (void)ws_size;
  const float* x_enc  = (const float*)d_in[0];
  const float* emb_w  = (const float*)d_in[1];
  const float* proj_w = (const float*)d_in[22];
  const float* proj_b = (const float*)d_in[23];
  const float* pred_w = (const float*)d_in[24];
  const float* pred_b = (const float*)d_in[25];

  const int NEL = 8 * 512 * 128;           // 524288 per-band working set
  float* ws    = (float*)d_ws;
  float* xr    = ws;                        // 4096
  float* bands = xr    + 4096;              // 4*8*512 = 16384
  float* fused = bands + 16384;             // 524288
  float* emb   = fused + NEL;               // 524288
  float* tx    = emb   + NEL;               // 524288
  float* tA    = tx    + NEL;               // 524288
  float* tB    = tA    + NEL;               // 524288
  float* h1    = tB    + NEL;               // 524288
  float* s0    = h1    + NEL;               // 128
  float* hbuf  = s0    + 128;               // 8*96*128 = 98304

  extract_kernel<<<16, 256, 0, stream>>>(x_enc, xr);
  dwt_bands_kernel<<<32, 128, 0, stream>>>(xr, bands);
  zero_kernel<<<2048, 256, 0, stream>>>(fused, NEL);

  const int Ks[4]   = {2, 3, 4, 5};
  const int dils[4] = {1, 2, 4, 8};
  for (int bi = 0; bi < 4; ++bi) {
    const float* cheb = (const float*)d_in[2 + bi * 5];
    const float* tw1  = (const float*)d_in[3 + bi * 5];
    const float* tb1  = (const float*)d_in[4 + bi * 5];
    const float* tw2  = (const float*)d_in[5 + bi * 5];
    const float* tb2  = (const float*)d_in[6 + bi * 5];
    const int K = Ks[bi], dil = dils[bi];

    embed_kernel<<<2048, 256, 0, stream>>>(bands + bi * 4096, emb_w, emb, tx);

    // Chebyshev: k=0 as column-sum bias fused into k=1 GEMM; k>=2 via recurrence
    cheb_colsum_kernel<<<1, 128, 0, stream>>>(cheb, K, s0);
    gemm128_acc_kernel<<<32, 256, 0, stream>>>(tx, cheb + 1, 128 * K, K, fused, s0);
    const float* Tlast = tx; const float* Tpp = nullptr;
    float* bufs[2] = {tA, tB};
    for (int k = 2; k < K; ++k) {
      float* Tnew = bufs[k & 1];
      cheb_rec_kernel<<<2048, 256, 0, stream>>>(tx, Tlast, Tpp, Tnew, NEL);
      gemm128_acc_kernel<<<32, 256, 0, stream>>>(Tnew, cheb + k, 128 * K, K, fused, nullptr);
      Tpp = Tlast; Tlast = Tnew;
    }

    // TCN: two dilated conv layers; second fuses residual(+emb) and accumulates into fused
    conv_gelu_kernel<<<32, 256, 0, stream>>>(emb, tw1, tb1, h1, nullptr, dil, 0);
    conv_gelu_kernel<<<32, 256, 0, stream>>>(h1, tw2, tb2, fused, emb, dil, 1);
  }

  pred_gemm_kernel<<<8, 256, 0, stream>>>(pred_w, fused, pred_b, hbuf);
  proj_kernel<<<9, 256, 0, stream>>>(hbuf, proj_w, proj_b, (float*)d_out);
}